// RocketTransform_33311766347805
// MI455X (gfx1250) — compile-verified
//
#include <hip/hip_runtime.h>

#define B_N 32
#define L_N 2048
#define K_N 1024
#define MAXLEN_N 11
#define NWAVES 4
#define ROWS_PER_WAVE (B_N / NWAVES)

// CDNA5 async global->LDS path (ASYNCcnt). Guarded so host pass / older
// toolchains fall back to float4 load + ds_store staging with prefetch.
#if defined(__gfx1250__) && __has_builtin(__builtin_amdgcn_global_load_async_to_lds_b128)
#define USE_ASYNC 1
#else
#define USE_ASYNC 0
#endif

#if USE_ASYNC
// Builtin signature (probe-confirmed by round-1 diagnostic): param0 is a
// non-const 'int __attribute__((vector_size(16)))' pointer in AS1.
typedef int v4i __attribute__((vector_size(16)));
typedef __attribute__((address_space(1))) v4i GV4;  // global int4
typedef __attribute__((address_space(3))) v4i LV4;  // LDS int4
#if __has_builtin(__builtin_amdgcn_s_wait_asynccnt)
#define WAIT_ASYNC(n) __builtin_amdgcn_s_wait_asynccnt(n)
#else
#define WAIT_ASYNC(n) asm volatile("s_wait_asynccnt " #n)
#endif
#endif

__device__ __forceinline__ void stage_row_async(const float* __restrict__ src,
                                                float* __restrict__ dst,
                                                int lane) {
#if USE_ASYNC
  // 16 async b128 ops sharing one global address pair + one LDS address VGPR;
  // the 24-bit instruction IOFFSET advances BOTH addresses (ISA 10.x):
  // 32 lanes x 16B = 512B per op -> 8KB row in 16 ops.
  GV4* g = (GV4*)(src + lane * 4);
  LV4* l = (LV4*)(dst + lane * 4);
#define STAGE1(OFF) __builtin_amdgcn_global_load_async_to_lds_b128(g, l, OFF, 0)
  STAGE1(0);    STAGE1(512);  STAGE1(1024); STAGE1(1536);
  STAGE1(2048); STAGE1(2560); STAGE1(3072); STAGE1(3584);
  STAGE1(4096); STAGE1(4608); STAGE1(5120); STAGE1(5632);
  STAGE1(6144); STAGE1(6656); STAGE1(7168); STAGE1(7680);
#undef STAGE1
#endif
}

__global__ __launch_bounds__(NWAVES * 32)
void rocket_transform_kernel(const float* __restrict__ X,
                             const float* __restrict__ W,
                             const float* __restrict__ bias,
                             const int* __restrict__ lengths,
                             const int* __restrict__ dilations,
                             const int* __restrict__ paddings,
                             float* __restrict__ out) {
  // 64 KB: per-wave double-buffered X row (conflict-free lane->bank mapping)
  __shared__ float lds[NWAVES][2][L_N];

  const int k = blockIdx.x;
  const int tid = threadIdx.x;
  const int wave = tid >> 5;
  const int lane = tid & 31;

  // Block-uniform kernel metadata -> scalar registers
  const int len = lengths[k];
  const int d = dilations[k];
  const int p = paddings[k];
  const float bk = bias[k];
  const int out_len = L_N + 2 * p - (len - 1) * d;
  const float inv_ol = 1.0f / (float)out_len;

  const float* wp = W + k * MAXLEN_N;
  float wreg[MAXLEN_N];
#pragma unroll
  for (int j = 0; j < MAXLEN_N; ++j) wreg[j] = wp[j]; // zero past len -> exact

  float* buf0 = &lds[wave][0][0];
  float* buf1 = &lds[wave][1][0];

  const int b0 = wave * ROWS_PER_WAVE;

#if USE_ASYNC
  stage_row_async(X + (size_t)b0 * L_N, buf0, lane);
#endif

  for (int r = 0; r < ROWS_PER_WAVE; ++r) {
    const int b = b0 + r;
    float* row;

#if USE_ASYNC
    row = (r & 1) ? buf1 : buf0;
    if (r + 1 < ROWS_PER_WAVE) {
      // issue next row into the other buffer, then wait for ours: async ops
      // complete in order, so <=16 outstanding => row r is resident.
      stage_row_async(X + (size_t)(b + 1) * L_N, (r & 1) ? buf0 : buf1, lane);
      WAIT_ASYNC(16);
    } else {
      WAIT_ASYNC(0);
    }
    // warm L2/WGP$ for the row after next (global_prefetch_b8)
    if (r + 2 < ROWS_PER_WAVE)
      __builtin_prefetch(X + (size_t)(b + 2) * L_N + lane * 64, 0, 0);
#else
    row = buf0;
    if (r + 1 < ROWS_PER_WAVE)
      __builtin_prefetch(X + (size_t)(b + 1) * L_N + lane * 64, 0, 0);
    {
      const float4* src = (const float4*)(X + (size_t)b * L_N);
      float4* dst = (float4*)row;
#pragma unroll
      for (int i = 0; i < 16; ++i) dst[i * 32 + lane] = src[i * 32 + lane];
      // same-wave LDS ops are in-order: ds_load after ds_store is safe
    }
#endif

    int cnt = 0;
    float mx = -__builtin_inff();

    for (int t0 = 0; t0 < L_N; t0 += 32) {
      const int t = t0 + lane;
      float conv = bk;
#pragma unroll
      for (int j = 0; j < MAXLEN_N; ++j) {
        const int base = t0 - p + j * d; // wave-uniform per chunk
        float xv;
        if (base >= 0 && base + 31 < L_N) {
          // interior fast path: ds_load + fma only
          xv = row[base + lane];
        } else if (base + 31 < 0 || base >= L_N) {
          xv = 0.0f; // whole chunk out of range
        } else {
          const unsigned u = (unsigned)(base + lane);
          const float v = row[u & (unsigned)(L_N - 1)]; // always in-bounds
          xv = (u < (unsigned)L_N) ? v : 0.0f;
        }
        conv = fmaf(wreg[j], xv, conv);
      }
      if (t < out_len) {
        cnt += (conv > 0.0f) ? 1 : 0;
        mx = fmaxf(mx, conv);
      }
    }

    // wave32 reductions
#pragma unroll
    for (int s = 16; s > 0; s >>= 1) {
      cnt += __shfl_xor(cnt, s, 32);
      mx = fmaxf(mx, __shfl_xor(mx, s, 32));
    }

    if (lane == 0) {
      float* o = out + (size_t)b * (2 * K_N) + 2 * k;
      o[0] = (float)cnt * inv_ol; // PPV
      o[1] = mx;                  // max
    }
  }
}

extern "C" void kernel_launch(void* const* d_in, const int* in_sizes, int n_in,
                              void* d_out, int out_size, void* d_ws, size_t ws_size,
                              hipStream_t stream) {
  (void)in_sizes; (void)n_in; (void)out_size; (void)d_ws; (void)ws_size;
  const float* X       = (const float*)d_in[0];
  const float* W       = (const float*)d_in[1];
  const float* bias    = (const float*)d_in[2];
  const int* lengths   = (const int*)d_in[3];
  const int* dilations = (const int*)d_in[4];
  const int* paddings  = (const int*)d_in[5];
  float* out = (float*)d_out;

  hipLaunchKernelGGL(rocket_transform_kernel, dim3(K_N), dim3(NWAVES * 32), 0,
                     stream, X, W, bias, lengths, dilations, paddings, out);
}